// YoloDecoderLayer_6399501271637
// MI455X (gfx1250) — compile-verified
//
#include <hip/hip_runtime.h>
#include <hip/hip_bf16.h>

#define NCLASSES 80
#define MAX_BOXES 100
#define IOU_THR 0.5f
#define NEG_SCORE (-1e9f)

#define N0 (48*48*3)            /* 6912   */
#define N1 (96*96*3)            /* 27648  */
#define N2 (192*192*3)          /* 110592 */
#define NTOT (N0+N1+N2)         /* 145152 */

#define DEC_BPB 128                         /* boxes per decode block   */
#define DEC_BLOCKS (NTOT / DEC_BPB)         /* 1134                     */
#define DEC_BLOCKS0 (N0 / DEC_BPB)          /* 54                       */
#define DEC_BLOCKS1 (N1 / DEC_BPB)          /* 216                      */
#define SLICE_ELEMS (32 * 85)               /* 2720 f32 per wave slice  */

typedef __attribute__((ext_vector_type(4))) unsigned int u32x4;
typedef __attribute__((ext_vector_type(4))) int          i32x4;
typedef __attribute__((ext_vector_type(8))) int          i32x8;

// ---------------------------------------------------------------------------
// Stage one 2720-float slice (32 boxes x 85) from global into LDS.
// CDNA5 path: Tensor Data Mover (tensor_load_to_lds) + s_wait_tensorcnt.
// Each wave issues its own descriptor (TDM ignores EXEC; one op per wave).
// This toolchain exposes the 6-arg builtin: (v4u, v8i, v4i, v4i, v8i, cpol).
// ---------------------------------------------------------------------------
__device__ __forceinline__ void stage_slice(float* lds_dst, const float* gsrc, int lane) {
#if defined(__has_builtin) && __has_builtin(__builtin_amdgcn_tensor_load_to_lds) && __has_builtin(__builtin_amdgcn_s_wait_tensorcnt)
  unsigned long long ga = (unsigned long long)(uintptr_t)gsrc;
  unsigned lds_off = (unsigned)(uintptr_t)lds_dst;   // low 32b of flat LDS addr = LDS offset
  constexpr unsigned TD0 = (unsigned)SLICE_ELEMS;    // 2720 elements, 4B each

  u32x4 g0;
  g0[0] = 1u;                                              // count=1, user descriptor
  g0[1] = lds_off;                                         // lds_addr (bytes)
  g0[2] = (unsigned)(ga & 0xFFFFFFFFull);                  // global_addr[31:0]
  g0[3] = (unsigned)((ga >> 32) & 0x01FFFFFFull)           // global_addr[56:32]
        | (2u << 30);                                      // type=2 ("image")

  i32x8 g1;
  g1[0] = (int)(2u << 16);                                 // wg_mask=0, data_size=2 (4B)
  g1[1] = (int)((TD0 & 0xFFFFu) << 16);                    // tensor_dim0[15:0]
  g1[2] = (int)(((TD0 >> 16) & 0xFFFFu) | (1u << 16));     // tensor_dim0[31:16], tensor_dim1=1
  g1[3] = (int)((TD0 & 0xFFFFu) << 16);                    // tile_dim0 = 2720
  g1[4] = 0;                                               // tile_dim1=0, tile_dim2=0 (unused)
  g1[5] = (int)TD0;                                        // tensor_dim0_stride[31:0]
  g1[6] = 0;
  g1[7] = 0;

  i32x4 z4; z4[0]=0; z4[1]=0; z4[2]=0; z4[3]=0;            // groups 2/3 unused (<=2D)
  i32x8 z8; z8[0]=0; z8[1]=0; z8[2]=0; z8[3]=0;
  z8[4]=0; z8[5]=0; z8[6]=0; z8[7]=0;                      // extra group (unused), zero-fill

  __builtin_amdgcn_tensor_load_to_lds(g0, g1, z4, z4, z8, 0);
  __builtin_amdgcn_s_wait_tensorcnt(0);
  (void)lane;
#else
  for (int e = lane; e < SLICE_ELEMS; e += 32) lds_dst[e] = gsrc[e];
#endif
}

// ---------------------------------------------------------------------------
// Kernel 1: decode boxes, conf, class-argmax; per-block partial max/min of
// per-box maxprob.
// ---------------------------------------------------------------------------
__global__ __launch_bounds__(DEC_BPB) void yolo_decode_kernel(
    const float* __restrict__ gr0, const float* __restrict__ gr1,
    const float* __restrict__ gr2, const float* __restrict__ anchors,
    float4* __restrict__ boxes, float* __restrict__ conf,
    int* __restrict__ cls, float* __restrict__ pmax, float* __restrict__ pmin)
{
  __shared__ float tile[DEC_BPB * 85];   // 43.5 KB
  __shared__ float redA[DEC_BPB];
  __shared__ float redB[DEC_BPB];

  const int b = blockIdx.x;
  const float* src; int gsel, base_box, S, gbase;
  if (b < DEC_BLOCKS0)                { gsel = 0; src = gr0; base_box = b * DEC_BPB;                           S = 48;  gbase = 0;      }
  else if (b < DEC_BLOCKS0+DEC_BLOCKS1){ gsel = 1; src = gr1; base_box = (b - DEC_BLOCKS0) * DEC_BPB;          S = 96;  gbase = N0;     }
  else                                { gsel = 2; src = gr2; base_box = (b - DEC_BLOCKS0 - DEC_BLOCKS1) * DEC_BPB; S = 192; gbase = N0+N1; }

  const int t    = threadIdx.x;
  const int lane = t & 31;
  const int w    = t >> 5;

  // Each of 4 waves DMA-stages its 32-box slice via TDM, waits TENSORcnt.
  stage_slice(tile + w * SLICE_ELEMS,
              src + (size_t)(base_box + 32 * w) * 85, lane);
  __syncthreads();

  const float* p = tile + t * 85;        // stride 85 (odd) -> bank-conflict free
  const int local = base_box + t;        // box index within this grid
  const int a  = local % 3;
  const int hw = local / 3;
  const int xc = hw % S;                 // column
  const int yr = hw / S;                 // row
  const float inv = 1.0f / (float)S;

  const float cx = (p[0] + (float)xc) * inv;
  const float cy = (p[1] + (float)yr) * inv;
  const int arow = 2 - gsel;             // grid0 uses anchors_table[2], etc.
  const float ax = anchors[arow * 6 + a * 2 + 0];
  const float ay = anchors[arow * 6 + a * 2 + 1];
  const float bw = expf(p[2]) * ax;
  const float bh = expf(p[3]) * ay;

  const float cf = p[4];
  float mx = p[5]; int mi = 0;
  #pragma unroll
  for (int k = 1; k < NCLASSES; ++k) {
    float v = p[5 + k];
    if (v > mx) { mx = v; mi = k; }      // strict > keeps first occurrence
  }

  const int gi = gbase + local;
  boxes[gi] = make_float4(cx - 0.5f * bw, cy - 0.5f * bh,
                          cx + 0.5f * bw, cy + 0.5f * bh);
  conf[gi] = cf;
  cls[gi]  = mi;

  // block reduce max/min of maxprob
  redA[t] = mx; redB[t] = mx;
  __syncthreads();
  for (int s = DEC_BPB / 2; s > 0; s >>= 1) {
    if (t < s) {
      redA[t] = fmaxf(redA[t], redA[t + s]);
      redB[t] = fminf(redB[t], redB[t + s]);
    }
    __syncthreads();
  }
  if (t == 0) { pmax[b] = redA[0]; pmin[b] = redB[0]; }
}

// ---------------------------------------------------------------------------
// Kernel 2: reduce per-block partials -> global Mx / Mn of maxprob.
// ---------------------------------------------------------------------------
__global__ __launch_bounds__(256) void reduce_mxmn_kernel(
    const float* __restrict__ pmax, const float* __restrict__ pmin,
    float* __restrict__ mxmn, int n)
{
  __shared__ float ra[256], rb[256];
  const int t = threadIdx.x;
  float a = -3.0e38f, b = 3.0e38f;
  for (int i = t; i < n; i += 256) { a = fmaxf(a, pmax[i]); b = fminf(b, pmin[i]); }
  ra[t] = a; rb[t] = b;
  __syncthreads();
  for (int s = 128; s > 0; s >>= 1) {
    if (t < s) { ra[t] = fmaxf(ra[t], ra[t + s]); rb[t] = fminf(rb[t], rb[t + s]); }
    __syncthreads();
  }
  if (t == 0) { mxmn[0] = ra[0]; mxmn[1] = rb[0]; }
}

// ---------------------------------------------------------------------------
// Kernel 3: scores[i] = max(conf*Mx, conf*Mn)
// ---------------------------------------------------------------------------
__global__ __launch_bounds__(256) void score_kernel(
    const float* __restrict__ conf, const float* __restrict__ mxmn,
    float* __restrict__ scores)
{
  const int i = blockIdx.x * 256 + threadIdx.x;
  if (i < NTOT) {
    const float Mx = mxmn[0], Mn = mxmn[1];
    const float c = conf[i];
    scores[i] = fmaxf(c * Mx, c * Mn);
  }
}

// ---------------------------------------------------------------------------
// Kernel 4: greedy NMS, single 1024-thread (32-wave) block.
// Fused pass per round: suppress vs previous selection + argmax of updated
// scores in ONE L2 scan. wave32 shuffle + LDS argmax reduction.
// ---------------------------------------------------------------------------
__device__ __forceinline__ void block_argmax(float bv, int bi,
                                             float* swv, int* swi,
                                             float* selv, int* seli)
{
  const int t = threadIdx.x, lane = t & 31, wid = t >> 5;
  #pragma unroll
  for (int o = 16; o > 0; o >>= 1) {
    float ov = __shfl_down(bv, o, 32);
    int   oi = __shfl_down(bi, o, 32);
    if (ov > bv || (ov == bv && oi < bi)) { bv = ov; bi = oi; }
  }
  if (lane == 0) { swv[wid] = bv; swi[wid] = bi; }
  __syncthreads();
  if (wid == 0) {
    bv = swv[lane]; bi = swi[lane];
    #pragma unroll
    for (int o = 16; o > 0; o >>= 1) {
      float ov = __shfl_down(bv, o, 32);
      int   oi = __shfl_down(bi, o, 32);
      if (ov > bv || (ov == bv && oi < bi)) { bv = ov; bi = oi; }
    }
    if (lane == 0) { *selv = bv; *seli = bi; }
  }
  __syncthreads();
}

__global__ __launch_bounds__(1024) void nms_kernel(
    const float4* __restrict__ boxes, float* __restrict__ scores,
    const int* __restrict__ cls, float* __restrict__ out, int out_n)
{
  __shared__ float swv[32];
  __shared__ int   swi[32];
  __shared__ float s_selv;
  __shared__ int   s_seli;
  __shared__ float s_box[4];
  __shared__ float s_area;
  __shared__ int   s_valid;

  const int t = threadIdx.x;

  // zero all outputs (invalid slots must be exactly 0, per reference)
  for (int i = t; i < out_n; i += 1024) out[i] = 0.0f;

  // initial argmax over scores
  float bv = -3.0e38f; int bi = 0x7FFFFFFF;
  for (int i = t; i < NTOT; i += 1024) {
    float s = scores[i];
    if (s > bv) { bv = s; bi = i; }
  }
  block_argmax(bv, bi, swv, swi, &s_selv, &s_seli);

  int nv = 0;  // meaningful on thread 0 only
  for (int it = 0; it < MAX_BOXES; ++it) {
    if (t == 0) {
      const float s = s_selv; const int j = s_seli;
      const int valid = (s > 0.0f) ? 1 : 0;
      s_valid = valid;
      if (valid) {
        const float4 bj = boxes[j];
        out[it * 4 + 0] = bj.x; out[it * 4 + 1] = bj.y;
        out[it * 4 + 2] = bj.z; out[it * 4 + 3] = bj.w;
        out[400 + it] = s;
        out[500 + it] = (float)cls[j];
        s_box[0] = bj.x; s_box[1] = bj.y; s_box[2] = bj.z; s_box[3] = bj.w;
        s_area = (bj.z - bj.x) * (bj.w - bj.y);
        ++nv;
      }
    }
    __syncthreads();
    if (!s_valid) break;               // scores frozen -> all remaining invalid (zeros)
    if (it == MAX_BOXES - 1) break;

    const float x1 = s_box[0], y1 = s_box[1], x2 = s_box[2], y2 = s_box[3];
    const float aj = s_area;

    bv = -3.0e38f; bi = 0x7FFFFFFF;
    for (int i = t; i < NTOT; i += 1024) {
      if (i + 8192 < NTOT) __builtin_prefetch(&scores[i + 8192], 0, 0);
      float s = scores[i];
      if (s > -1.0e8f) {               // skip already-suppressed (stays NEG either way)
        const float4 bb = boxes[i];
        const float iw = fmaxf(fminf(bb.z, x2) - fmaxf(bb.x, x1), 0.0f);
        const float ih = fmaxf(fminf(bb.w, y2) - fmaxf(bb.y, y1), 0.0f);
        const float inter = iw * ih;
        const float ai = (bb.z - bb.x) * (bb.w - bb.y);
        const float iou = inter / (ai + aj - inter);
        if (iou > IOU_THR) { s = NEG_SCORE; scores[i] = NEG_SCORE; }
      }
      if (s > bv) { bv = s; bi = i; }
    }
    // block_argmax's internal barriers order the global score writes
    block_argmax(bv, bi, swv, swi, &s_selv, &s_seli);
  }
  if (t == 0) out[600] = (float)nv;
}

// ---------------------------------------------------------------------------
extern "C" void kernel_launch(void* const* d_in, const int* in_sizes, int n_in,
                              void* d_out, int out_size, void* d_ws, size_t ws_size,
                              hipStream_t stream) {
  (void)in_sizes; (void)n_in; (void)ws_size;
  const float* gr0  = (const float*)d_in[0];
  const float* gr1  = (const float*)d_in[1];
  const float* gr2  = (const float*)d_in[2];
  const float* anch = (const float*)d_in[3];
  float* out = (float*)d_out;

  char* ws = (char*)d_ws;
  float4* boxes  = (float4*)ws;               ws += (size_t)NTOT * sizeof(float4);
  float*  conf   = (float*)ws;                ws += (size_t)NTOT * sizeof(float);
  float*  scores = (float*)ws;                ws += (size_t)NTOT * sizeof(float);
  int*    cls    = (int*)ws;                  ws += (size_t)NTOT * sizeof(int);
  float*  pmax   = (float*)ws;                ws += (size_t)DEC_BLOCKS * sizeof(float);
  float*  pmin   = (float*)ws;                ws += (size_t)DEC_BLOCKS * sizeof(float);
  float*  mxmn   = (float*)ws;                /* 2 floats */

  yolo_decode_kernel<<<DEC_BLOCKS, DEC_BPB, 0, stream>>>(
      gr0, gr1, gr2, anch, boxes, conf, cls, pmax, pmin);
  reduce_mxmn_kernel<<<1, 256, 0, stream>>>(pmax, pmin, mxmn, DEC_BLOCKS);
  score_kernel<<<NTOT / 256, 256, 0, stream>>>(conf, mxmn, scores);
  nms_kernel<<<1, 1024, 0, stream>>>(boxes, scores, cls, out, out_size);
}